// MultiBoxLoss_25890062860671
// MI455X (gfx1250) — compile-verified
//
#include <hip/hip_runtime.h>
#include <cstdint>

// ---- problem constants (match reference) ----
#define NB 32
#define NP 65536
#define NO 32
#define THRESH_F 0.35f
#define VAR0 0.1f
#define VAR1 0.2f
#define LOC_W 2.0f
#define LANDM_W 1.0f
#define ALPHA_F 0.25f
#define EPS_F 1e-7f

#define TPB 256
#define PRB 4                      // priors per thread
#define BLKX (NP / (TPB * PRB))    // 64 blocks along priors
#define NBLK (BLKX * NB)           // 2048 partial-sum blocks

typedef __attribute__((ext_vector_type(2))) float v2f;
typedef __attribute__((ext_vector_type(8))) float v8f;

__device__ __forceinline__ float smooth_l1f(float x) {
  float a = fabsf(x);
  return a < 1.0f ? 0.5f * x * x : a - 0.5f;
}

__device__ __forceinline__ unsigned long long shfl_xor_u64(unsigned long long x, int m) {
  unsigned lo = __shfl_xor((unsigned)x, m, 32);
  unsigned hi = __shfl_xor((unsigned)(x >> 32), m, 32);
  return ((unsigned long long)hi << 32) | (unsigned long long)lo;
}

// CDNA5 async global->LDS staging of targets[b] (480 floats = 120 x b128 lanes).
// LDS byte offset = low 32 bits of the generic shared pointer (flat LDS aperture).
__device__ __forceinline__ void stage_targets(const float* __restrict__ g, float* lds, int tid) {
  for (int i = tid; i < (NO * 15) / 4; i += TPB) {
    unsigned lo = (unsigned)(uintptr_t)(lds + i * 4);
    unsigned long long ga = (unsigned long long)(uintptr_t)(g + i * 4);
    asm volatile("global_load_async_to_lds_b128 %0, %1, off" : : "v"(lo), "v"(ga) : "memory");
  }
  asm volatile("s_wait_asynccnt 0" : : : "memory");
}

__global__ __launch_bounds__(TPB) void init_bpi_kernel(unsigned long long* bpi) {
  int i = blockIdx.x * TPB + threadIdx.x;
  if (i < NB * NO) bpi[i] = 0ull;
}

// Pass 1: per-(batch,truth) argmax over all priors of IoU.
// Packed key = (iou_bits << 32) | ~prior_idx  -> max picks larger IoU, then SMALLER
// prior index on ties (matches jnp.argmax first-index semantics). IoU >= 0 so float
// bit pattern is order-preserving.
__global__ __launch_bounds__(TPB) void match_kernel(const float4* __restrict__ priors,
                                                    const float* __restrict__ targets,
                                                    unsigned long long* __restrict__ bpi) {
  __shared__ __align__(16) float tgt[NO * 15];
  const int tid = threadIdx.x;
  const int b = blockIdx.y;
  stage_targets(targets + b * NO * 15, tgt, tid);
  __syncthreads();

  const int pbase = blockIdx.x * (TPB * PRB) + tid;
  float bx1[PRB], by1[PRB], bx2[PRB], by2[PRB], areaB[PRB];
#pragma unroll
  for (int i = 0; i < PRB; ++i) {
    float4 q = priors[pbase + i * TPB];
    bx1[i] = q.x - 0.5f * q.z; by1[i] = q.y - 0.5f * q.w;
    bx2[i] = q.x + 0.5f * q.z; by2[i] = q.y + 0.5f * q.w;
    areaB[i] = (bx2[i] - bx1[i]) * (by2[i] - by1[i]);
  }

  for (int t = 0; t < NO; ++t) {
    float ax1 = tgt[t*15+0], ay1 = tgt[t*15+1], ax2 = tgt[t*15+2], ay2 = tgt[t*15+3];
    float areaA = (ax2 - ax1) * (ay2 - ay1);
    unsigned long long best = 0ull;
#pragma unroll
    for (int i = 0; i < PRB; ++i) {
      float iw = fmaxf(fminf(ax2, bx2[i]) - fmaxf(ax1, bx1[i]), 0.0f);
      float ih = fmaxf(fminf(ay2, by2[i]) - fmaxf(ay1, by1[i]), 0.0f);
      float inter = iw * ih;
      float iou = inter / (areaA + areaB[i] - inter);
      unsigned long long pk = ((unsigned long long)__float_as_uint(iou) << 32)
                            | (unsigned long long)(~(unsigned)(pbase + i * TPB));
      best = pk > best ? pk : best;
    }
#pragma unroll
    for (int off = 16; off > 0; off >>= 1) {
      unsigned long long o = shfl_xor_u64(best, off);
      best = o > best ? o : best;
    }
    if ((tid & 31) == 0) atomicMax(&bpi[b * NO + t], best);
  }
}

// Pass 2: recompute per-prior best truth (cheap, L2-resident operands), apply the
// best-prior override, accumulate all three losses + positive count.
__global__ __launch_bounds__(TPB) void loss_kernel(const float4* __restrict__ loc4,
                                                   const float2* __restrict__ conf2,
                                                   const float* __restrict__ landm,
                                                   const float4* __restrict__ priors,
                                                   const float* __restrict__ targets,
                                                   const unsigned long long* __restrict__ bpi,
                                                   float* __restrict__ partials) {
  __shared__ __align__(16) float tgt[NO * 15];
  __shared__ unsigned bp_prior[NO];
  __shared__ float red[TPB / 32][4];
  const int tid = threadIdx.x;
  const int b = blockIdx.y;
  stage_targets(targets + b * NO * 15, tgt, tid);
  if (tid < NO) bp_prior[tid] = ~(unsigned)bpi[b * NO + tid];  // recover prior index
  __syncthreads();

  float ll = 0.0f, lc = 0.0f, llm = 0.0f, n1 = 0.0f;
  const int pbase = blockIdx.x * (TPB * PRB) + tid;
#pragma unroll 1
  for (int i = 0; i < PRB; ++i) {
    const int p = pbase + i * TPB;
    float4 q = priors[p];
    const float cx = q.x, cy = q.y, w = q.z, h = q.w;
    const float bx1 = cx - 0.5f*w, by1 = cy - 0.5f*h, bx2 = cx + 0.5f*w, by2 = cy + 0.5f*h;
    const float areaB = (bx2 - bx1) * (by2 - by1);

    float bto = -1.0f; int bti = 0;                 // strict '>' keeps first max (jnp.argmax)
    for (int t = 0; t < NO; ++t) {
      float ax1 = tgt[t*15+0], ay1 = tgt[t*15+1], ax2 = tgt[t*15+2], ay2 = tgt[t*15+3];
      float areaA = (ax2 - ax1) * (ay2 - ay1);
      float iw = fmaxf(fminf(ax2, bx2) - fmaxf(ax1, bx1), 0.0f);
      float ih = fmaxf(fminf(ay2, by2) - fmaxf(ay1, by1), 0.0f);
      float inter = iw * ih;
      float iou = inter / (areaA + areaB - inter);
      if (iou > bto) { bto = iou; bti = t; }
    }
#pragma unroll
    for (int t = 0; t < NO; ++t)                    // ascending: last write wins (.at[].set)
      if (bp_prior[t] == (unsigned)p) { bto = 2.0f; bti = t; }

    const float label = tgt[bti*15 + 14];
    const float conf_raw = (bto < THRESH_F) ? 0.0f : label;
    const bool pos  = conf_raw != 0.0f;
    const bool pos1 = conf_raw > 0.0f;

    const long long idx = (long long)b * NP + p;

    // focal loss over C=2 classes (always evaluated)
    float2 cd = conf2[idx];
    float mm = fmaxf(cd.x, cd.y);
    float lse = mm + logf(expf(cd.x - mm) + expf(cd.y - mm));
    float logpt = (pos ? cd.y : cd.x) - lse;
    float pt = expf(logpt);
    float om = fmaxf(1.0f - pt, 0.0f);
    lc += -ALPHA_F * om * sqrtf(om) * logpt;        // (1-pt)^1.5

    if (pos) {                                      // GIoU loc loss
      float4 l = loc4[idx];
      float dcx = cx + l.x * VAR0 * w;
      float dcy = cy + l.y * VAR0 * h;
      float dw = w * expf(l.z * VAR1);
      float dh = h * expf(l.w * VAR1);
      float d1 = dcx - 0.5f*dw, d2 = dcy - 0.5f*dh, d3 = dcx + 0.5f*dw, d4 = dcy + 0.5f*dh;
      float tx1 = tgt[bti*15+0], ty1 = tgt[bti*15+1], tx2 = tgt[bti*15+2], ty2 = tgt[bti*15+3];
      float area1 = (d3 - d1) * (d4 - d2);
      float area2 = (tx2 - tx1) * (ty2 - ty1);
      float iw = fmaxf(fminf(d3, tx2) - fmaxf(d1, tx1), 0.0f);
      float ih = fmaxf(fminf(d4, ty2) - fmaxf(d2, ty1), 0.0f);
      float inter = iw * ih;
      float uni = area1 + area2 - inter;
      float iou = inter / (uni + EPS_F);
      float ew = fmaxf(d3, tx2) - fminf(d1, tx1);
      float eh = fmaxf(d4, ty2) - fminf(d2, ty1);
      float enc = ew * eh;
      ll += 1.0f - (iou - (enc - uni) / (enc + EPS_F));
    }
    if (pos1) {                                     // landmark smooth-L1
      const float* ld = landm + idx * 10;
      float inv_w = 1.0f / (VAR0 * w);
      float inv_h = 1.0f / (VAR0 * h);
      float s = 0.0f;
#pragma unroll
      for (int j = 0; j < 5; ++j) {
        float gx = (tgt[bti*15+4+2*j] - cx) * inv_w;
        float gy = (tgt[bti*15+5+2*j] - cy) * inv_h;
        s += smooth_l1f(ld[2*j] - gx) + smooth_l1f(ld[2*j+1] - gy);
      }
      llm += s;
      n1 += 1.0f;
    }
  }

  // deterministic block reduction: shuffles -> LDS -> thread 0 -> per-block partial
#pragma unroll
  for (int off = 16; off > 0; off >>= 1) {
    ll  += __shfl_xor(ll,  off, 32);
    lc  += __shfl_xor(lc,  off, 32);
    llm += __shfl_xor(llm, off, 32);
    n1  += __shfl_xor(n1,  off, 32);
  }
  if ((tid & 31) == 0) {
    red[tid >> 5][0] = ll;  red[tid >> 5][1] = lc;
    red[tid >> 5][2] = llm; red[tid >> 5][3] = n1;
  }
  __syncthreads();
  if (tid == 0) {
    float s0 = 0, s1 = 0, s2 = 0, s3 = 0;
    for (int wv = 0; wv < TPB / 32; ++wv) { s0 += red[wv][0]; s1 += red[wv][1]; s2 += red[wv][2]; s3 += red[wv][3]; }
    int blk = blockIdx.y * gridDim.x + blockIdx.x;
    ((float4*)partials)[blk] = make_float4(s0, s1, s2, s3);
  }
}

// WMMA-based exact fp32 32-lane sum: A VGPR0 = v (K=0 / K=2 slots), VGPR1 = 0,
// B = ones -> D[m][n] = v(m) + v(m+16) for every n. Lane L<16 holds D[r][L] in d[r],
// lane L>=16 holds D[r+8][L-16]; summing a lane's 8 D regs + xor-16 shuffle gives the
// full column sum (= grand total) in every lane.
#if __has_builtin(__builtin_amdgcn_wmma_f32_16x16x4_f32)
#define HAVE_WMMA_X4 1
#else
#define HAVE_WMMA_X4 0
#endif

__device__ __forceinline__ float wave32_sum(float v) {
#if HAVE_WMMA_X4
  v2f a; a.x = v; a.y = 0.0f;
  v2f bo; bo.x = 1.0f; bo.y = 1.0f;
  v8f c = {};
  v8f d = __builtin_amdgcn_wmma_f32_16x16x4_f32(false, a, false, bo, (short)0, c, false, false);
  float s = d[0] + d[1] + d[2] + d[3] + d[4] + d[5] + d[6] + d[7];
  s += __shfl_xor(s, 16, 32);
  return s;
#else
#pragma unroll
  for (int off = 16; off > 0; off >>= 1) v += __shfl_xor(v, off, 32);
  return v;
#endif
}

__global__ __launch_bounds__(TPB) void final_kernel(const float* __restrict__ partials,
                                                    float* __restrict__ out) {
  __shared__ float red[TPB / 32][4];
  const int tid = threadIdx.x;
  float s0 = 0, s1 = 0, s2 = 0, s3 = 0;
  for (int i = tid; i < NBLK; i += TPB) {
    float4 v = ((const float4*)partials)[i];
    s0 += v.x; s1 += v.y; s2 += v.z; s3 += v.w;
  }
#pragma unroll
  for (int off = 16; off > 0; off >>= 1) {
    s0 += __shfl_xor(s0, off, 32);
    s1 += __shfl_xor(s1, off, 32);
    s2 += __shfl_xor(s2, off, 32);
    s3 += __shfl_xor(s3, off, 32);
  }
  if ((tid & 31) == 0) { red[tid>>5][0]=s0; red[tid>>5][1]=s1; red[tid>>5][2]=s2; red[tid>>5][3]=s3; }
  __syncthreads();
  if (tid < 32) {                      // wave 0 fully active -> EXEC all-ones for WMMA
    float v0 = (tid < TPB/32) ? red[tid][0] : 0.0f;
    float v1 = (tid < TPB/32) ? red[tid][1] : 0.0f;
    float v2 = (tid < TPB/32) ? red[tid][2] : 0.0f;
    float v3 = (tid < TPB/32) ? red[tid][3] : 0.0f;
    float t0 = wave32_sum(v0);
    float t1 = wave32_sum(v1);
    float t2 = wave32_sum(v2);
    float t3 = wave32_sum(v3);
    if (tid == 0) {
      float N1 = fmaxf(t3, 1.0f);
      out[0] = LOC_W   * t0 / N1;
      out[1] =           t1 / N1;
      out[2] = LANDM_W * t2 / N1;
    }
  }
}

extern "C" void kernel_launch(void* const* d_in, const int* in_sizes, int n_in,
                              void* d_out, int out_size, void* d_ws, size_t ws_size,
                              hipStream_t stream) {
  (void)in_sizes; (void)n_in; (void)out_size; (void)ws_size;
  const float4* loc4    = (const float4*)d_in[0];   // (B,P,4)  f32
  const float2* conf2   = (const float2*)d_in[1];   // (B,P,2)  f32
  const float*  landm   = (const float*)d_in[2];    // (B,P,10) f32
  const float4* priors  = (const float4*)d_in[3];   // (P,4)    f32
  const float*  targets = (const float*)d_in[4];    // (B,O,15) f32
  float* out = (float*)d_out;

  // workspace: [0, 8KB) packed best-prior keys; [8KB, 40KB) per-block partial sums
  unsigned long long* bpi = (unsigned long long*)d_ws;
  float* partials = (float*)((char*)d_ws + (size_t)NB * NO * sizeof(unsigned long long));

  init_bpi_kernel<<<dim3((NB * NO + TPB - 1) / TPB), TPB, 0, stream>>>(bpi);
  match_kernel<<<dim3(BLKX, NB), TPB, 0, stream>>>(priors, targets, bpi);
  loss_kernel<<<dim3(BLKX, NB), TPB, 0, stream>>>(loc4, conf2, landm, priors, targets, bpi, partials);
  final_kernel<<<1, TPB, 0, stream>>>(partials, out);
}